// SEGNNConv_42700564856854
// MI455X (gfx1250) — compile-verified
//
#include <hip/hip_runtime.h>
#include <hip/hip_bf16.h>

// ---------------------------------------------------------------------------
// SEGNN equivariant conv for MI455X (gfx1250, wave32, WMMA).
// All dense contractions use V_WMMA_F32_16X16X4_F32 (fp32 matrix pipe).
// Edge MLP is fused into the message kernel (avoids a 410MB intermediate).
// Scatter-add uses hardware global_atomic_add_f32 (L2-resident accumulators).
// ---------------------------------------------------------------------------

typedef float v2f __attribute__((ext_vector_type(2)));
typedef float v8f __attribute__((ext_vector_type(8)));

#define C_ 64
#define A_ 16
#define B_ 8
#define H_ 16

#define INV_SQRT_C  0.125f                  // 1/sqrt(64)
#define INV_SQRT_A  0.25f                   // 1/sqrt(16)
#define INV_SQRT_H  0.25f                   // 1/sqrt(16)
#define INV_SQRT_B  0.35355339059327373f    // 1/sqrt(8)
#define INV_SQRT3   0.5773502691896258f
#define INV_SQRT2   0.7071067811865476f
#define INV_DEG     0.25f                   // 1/sqrt(16.0)
#define INV_SQRT_CA 0.03125f                // 1/sqrt(1024)

__device__ __forceinline__ v8f vzero8() {
  v8f z = {0.f, 0.f, 0.f, 0.f, 0.f, 0.f, 0.f, 0.f};
  return z;
}

// D = A(16x4 f32) * B(4x16 f32) + C(16x16 f32)
__device__ __forceinline__ v8f wmma4(v2f a, v2f b, v8f c) {
  return __builtin_amdgcn_wmma_f32_16x16x4_f32(false, a, false, b, (short)0, c,
                                               false, false);
}

// ---------------------------------------------------------------------------
// Kernel 1: s_pre = feats_s @ W1_s * invC ; v_pre[x] = feats_v[:,:,x] @ W1_v.
// Also zeroes the message accumulators for this node tile.
// 128 threads = 4 waves; wave w computes output columns [16w, 16w+16).
// ---------------------------------------------------------------------------
__global__ void node_pre_kernel(const float* __restrict__ fs,
                                const float* __restrict__ fv,
                                const float* __restrict__ W1s,
                                const float* __restrict__ W1v,
                                float* __restrict__ s_pre,
                                float* __restrict__ v_pre,
                                float* __restrict__ msg_s,
                                float* __restrict__ msg_v) {
  __shared__ float ldsX[16 * C_];
  const int t    = threadIdx.x;
  const int base = blockIdx.x * 16;

  // zero message accumulators for these 16 nodes
  for (int i = t; i < 16 * C_; i += 128) msg_s[base * C_ + i] = 0.0f;
  for (int i = t; i < 16 * C_ * 3; i += 128) msg_v[base * C_ * 3 + i] = 0.0f;

  const int lane    = t & 31;
  const int wave    = t >> 5;
  const int ln      = lane & 15;        // A row m / B-D col n
  const int kb      = (lane >> 4) * 2;  // A/B k base
  const int mb      = (lane >> 4) * 8;  // C/D row base
  const int colbase = wave * 16;

  for (int p = 0; p < 4; ++p) {
    __syncthreads();
    if (p == 0) {
      for (int i = t; i < 16 * C_; i += 128) ldsX[i] = fs[base * C_ + i];
    } else {
      const int x = p - 1;
      for (int i = t; i < 16 * C_; i += 128) {
        const int m = i >> 6, c = i & 63;
        ldsX[i] = fv[((base + m) * C_ + c) * 3 + x];  // (N,C,3) slice
      }
    }
    __syncthreads();

    const float* W = (p == 0) ? W1s : W1v;
    v8f acc = vzero8();
    for (int k0 = 0; k0 < C_; k0 += 4) {
      v2f a, b;
      a[0] = ldsX[ln * C_ + k0 + kb];
      a[1] = ldsX[ln * C_ + k0 + kb + 1];
      b[0] = W[(k0 + kb) * C_ + colbase + ln];
      b[1] = W[(k0 + kb + 1) * C_ + colbase + ln];
      acc  = wmma4(a, b, acc);
    }
    if (p == 0) {
#pragma unroll
      for (int j = 0; j < 8; ++j)
        s_pre[(base + mb + j) * C_ + colbase + ln] = acc[j] * INV_SQRT_C;
    } else {
      const int x = p - 1;
#pragma unroll
      for (int j = 0; j < 8; ++j)
        v_pre[((base + mb + j) * 3 + x) * C_ + colbase + ln] =
            acc[j] * INV_SQRT_C;
    }
  }
}

// ---------------------------------------------------------------------------
// Kernel 2: one wave per edge. Fused edge MLP (silu(emb@Wm1)@Wm2 -> w[5][C]),
// gather s/v at src, equivariant message, scatter-add into msg_* at dst.
// Lane l handles channels l and l+32.
// ---------------------------------------------------------------------------
__global__ void edge_kernel(const float* __restrict__ emb,
                            const float* __restrict__ sh,
                            const int* __restrict__ eidx, long E,
                            const float* __restrict__ Wm1,
                            const float* __restrict__ Wm2,
                            const float* __restrict__ s_pre,
                            const float* __restrict__ v_pre,
                            float* __restrict__ msg_s,
                            float* __restrict__ msg_v) {
  __shared__ float lWm1[B_ * H_];
  __shared__ float lWm2[H_ * 5 * C_];
  const int t = threadIdx.x;
  for (int i = t; i < B_ * H_; i += 256) lWm1[i] = Wm1[i];
  for (int i = t; i < H_ * 5 * C_; i += 256) lWm2[i] = Wm2[i];
  __syncthreads();

  const int lane = t & 31;
  const int wave = t >> 5;
  const long e   = (long)blockIdx.x * 8 + wave;
  if (e >= E) return;

  const int src = eidx[e];
  const int dst = eidx[E + e];

  // h[j] computed by lane with (lane&15)==j, broadcast via shuffles below
  float hacc       = 0.f;
  const float* er  = emb + e * B_;
#pragma unroll
  for (int b = 0; b < B_; ++b) hacc += er[b] * lWm1[b * H_ + (lane & 15)];
  hacc *= INV_SQRT_B;
  const float hv = hacc / (1.0f + __expf(-hacc));  // SiLU

  // w[k][c] for c = lane and lane+32
  float wa[5] = {0.f, 0.f, 0.f, 0.f, 0.f};
  float wb[5] = {0.f, 0.f, 0.f, 0.f, 0.f};
#pragma unroll
  for (int j = 0; j < H_; ++j) {
    const float hj   = __shfl(hv, j, 32);
    const float* row = &lWm2[j * 5 * C_];
#pragma unroll
    for (int k = 0; k < 5; ++k) {
      wa[k] += hj * row[k * C_ + lane];
      wb[k] += hj * row[k * C_ + lane + 32];
    }
  }
#pragma unroll
  for (int k = 0; k < 5; ++k) {
    wa[k] *= INV_SQRT_H;
    wb[k] *= INV_SQRT_H;
  }

  const float y0  = sh[e * 4 + 0];
  const float y1x = sh[e * 4 + 1];
  const float y1y = sh[e * 4 + 2];
  const float y1z = sh[e * 4 + 3];

#pragma unroll
  for (int half = 0; half < 2; ++half) {
    const int c    = lane + half * 32;
    const float w0 = half ? wb[0] : wa[0];
    const float w1 = half ? wb[1] : wa[1];
    const float w2 = half ? wb[2] : wa[2];
    const float w3 = half ? wb[3] : wa[3];
    const float w4 = half ? wb[4] : wa[4];

    const float se = s_pre[src * C_ + c];
    const float vx = v_pre[(src * 3 + 0) * C_ + c];
    const float vy = v_pre[(src * 3 + 1) * C_ + c];
    const float vz = v_pre[(src * 3 + 2) * C_ + c];

    const float dotv = vx * y1x + vy * y1y + vz * y1z;
    const float ms =
        INV_DEG * INV_SQRT2 * (w0 * se * y0 + INV_SQRT3 * w3 * dotv);

    const float cx = vy * y1z - vz * y1y;
    const float cy = vz * y1x - vx * y1z;
    const float cz = vx * y1y - vy * y1x;
    const float kv = INV_DEG * INV_SQRT3;
    const float mvx = kv * (w1 * se * y1x + w2 * vx * y0 + INV_SQRT2 * w4 * cx);
    const float mvy = kv * (w1 * se * y1y + w2 * vy * y0 + INV_SQRT2 * w4 * cy);
    const float mvz = kv * (w1 * se * y1z + w2 * vz * y0 + INV_SQRT2 * w4 * cz);

    unsafeAtomicAdd(&msg_s[dst * C_ + c], ms);
    unsafeAtomicAdd(&msg_v[(dst * 3 + 0) * C_ + c], mvx);
    unsafeAtomicAdd(&msg_v[(dst * 3 + 1) * C_ + c], mvy);
    unsafeAtomicAdd(&msg_v[(dst * 3 + 2) * C_ + c], mvz);
  }
}

// ---------------------------------------------------------------------------
// Kernel 3: per 16-node tile, per component p in {s, vx, vy, vz}:
//   t2 = msg@W2*invC ; tu = (attrs*invA)@Tu^T ; u = t2*tu
//   o  = u@W3*invC + (feats ⊗ attrs)@Wsc * invCA   (on-the-fly A fragments)
// Writes the final concatenated (N,256) output.
// ---------------------------------------------------------------------------
__global__ void node_post_kernel(
    const float* __restrict__ fs, const float* __restrict__ fv,
    const float* __restrict__ attrs, const float* __restrict__ msg_s,
    const float* __restrict__ msg_v, const float* __restrict__ W2s,
    const float* __restrict__ W2v, const float* __restrict__ Tus,
    const float* __restrict__ Tuv, const float* __restrict__ W3s,
    const float* __restrict__ W3v, const float* __restrict__ Wscs,
    const float* __restrict__ Wscv, float* __restrict__ out) {
  __shared__ float ldsA[16 * C_];   // msg tile, then u staging
  __shared__ float ldsF[16 * C_];   // feats tile (scaled by invCA)
  __shared__ float ldsAt[16 * A_];  // raw attrs tile

  const int t       = threadIdx.x;
  const int base    = blockIdx.x * 16;
  const int lane    = t & 31;
  const int wave    = t >> 5;
  const int ln      = lane & 15;
  const int kb      = (lane >> 4) * 2;
  const int mb      = (lane >> 4) * 8;
  const int colbase = wave * 16;

  for (int i = t; i < 16 * A_; i += 128) ldsAt[i] = attrs[base * A_ + i];

  for (int p = 0; p < 4; ++p) {
    __syncthreads();
    if (p == 0) {
      for (int i = t; i < 16 * C_; i += 128) {
        ldsA[i] = msg_s[base * C_ + i];
        ldsF[i] = fs[base * C_ + i] * INV_SQRT_CA;
      }
    } else {
      const int x = p - 1;
      for (int i = t; i < 16 * C_; i += 128) {
        const int m = i >> 6, c = i & 63;
        ldsA[i] = msg_v[((base + m) * 3 + x) * C_ + c];
        ldsF[i] = fv[((base + m) * C_ + c) * 3 + x] * INV_SQRT_CA;
      }
    }
    __syncthreads();

    const float* W2  = (p == 0) ? W2s : W2v;
    const float* Tu  = (p == 0) ? Tus : Tuv;
    const float* W3  = (p == 0) ? W3s : W3v;
    const float* Wsc = (p == 0) ? Wscs : Wscv;

    // t2 = msg @ W2 (scale applied later)
    v8f t2 = vzero8();
    for (int k0 = 0; k0 < C_; k0 += 4) {
      v2f a, b;
      a[0] = ldsA[ln * C_ + k0 + kb];
      a[1] = ldsA[ln * C_ + k0 + kb + 1];
      b[0] = W2[(k0 + kb) * C_ + colbase + ln];
      b[1] = W2[(k0 + kb + 1) * C_ + colbase + ln];
      t2   = wmma4(a, b, t2);
    }

    // tu = (attrs * invA) @ Tu^T   ; Tu is (C,A), B[k][n] = Tu[(col+n)*A + k]
    v8f tu = vzero8();
    for (int k0 = 0; k0 < A_; k0 += 4) {
      v2f a, b;
      a[0] = ldsAt[ln * A_ + k0 + kb] * INV_SQRT_A;
      a[1] = ldsAt[ln * A_ + k0 + kb + 1] * INV_SQRT_A;
      b[0] = Tu[(colbase + ln) * A_ + k0 + kb];
      b[1] = Tu[(colbase + ln) * A_ + k0 + kb + 1];
      tu   = wmma4(a, b, tu);
    }

    // u = (t2 * invC) * tu  — identical C/D layouts so elementwise is exact
    v8f u;
#pragma unroll
    for (int j = 0; j < 8; ++j) u[j] = t2[j] * INV_SQRT_C * tu[j];

    __syncthreads();  // all waves done reading msg tile
#pragma unroll
    for (int j = 0; j < 8; ++j) ldsA[(mb + j) * C_ + colbase + ln] = u[j];
    __syncthreads();

    // o = u @ W3 * invC
    v8f o = vzero8();
    for (int k0 = 0; k0 < C_; k0 += 4) {
      v2f a, b;
      a[0] = ldsA[ln * C_ + k0 + kb];
      a[1] = ldsA[ln * C_ + k0 + kb + 1];
      b[0] = W3[(k0 + kb) * C_ + colbase + ln];
      b[1] = W3[(k0 + kb + 1) * C_ + colbase + ln];
      o    = wmma4(a, b, o);
    }
#pragma unroll
    for (int j = 0; j < 8; ++j) o[j] *= INV_SQRT_C;

    // o += (feats ⊗ attrs) @ Wsc ; K = C*A = 1024, A built on the fly.
    // k = u*16 + v ; A[m][k] = ldsF[m][k>>4] * ldsAt[m][k&15]
    for (int k0 = 0; k0 < C_ * A_; k0 += 4) {
      const int k1 = k0 + kb, k2 = k1 + 1;
      v2f a, b;
      a[0] = ldsF[ln * C_ + (k1 >> 4)] * ldsAt[ln * A_ + (k1 & 15)];
      a[1] = ldsF[ln * C_ + (k2 >> 4)] * ldsAt[ln * A_ + (k2 & 15)];
      b[0] = Wsc[k1 * C_ + colbase + ln];
      b[1] = Wsc[k2 * C_ + colbase + ln];
      o    = wmma4(a, b, o);
    }

    if (p == 0) {
#pragma unroll
      for (int j = 0; j < 8; ++j)
        out[(base + mb + j) * 256 + colbase + ln] = o[j];
    } else {
      const int x = p - 1;
#pragma unroll
      for (int j = 0; j < 8; ++j)
        out[(base + mb + j) * 256 + C_ + (colbase + ln) * 3 + x] = o[j];
    }
  }
}

// ---------------------------------------------------------------------------
extern "C" void kernel_launch(void* const* d_in, const int* in_sizes, int n_in,
                              void* d_out, int out_size, void* d_ws,
                              size_t ws_size, hipStream_t stream) {
  const float* fs    = (const float*)d_in[0];
  const float* fv    = (const float*)d_in[1];
  const float* attrs = (const float*)d_in[2];
  const float* emb   = (const float*)d_in[3];
  const float* sh    = (const float*)d_in[4];
  const int* eidx    = (const int*)d_in[5];
  const float* W1s   = (const float*)d_in[6];
  const float* W1v   = (const float*)d_in[7];
  const float* Wm1   = (const float*)d_in[8];
  const float* Wm2   = (const float*)d_in[9];
  const float* W2s   = (const float*)d_in[10];
  const float* W2v   = (const float*)d_in[11];
  const float* Tus   = (const float*)d_in[12];
  const float* Tuv   = (const float*)d_in[13];
  const float* W3s   = (const float*)d_in[14];
  const float* W3v   = (const float*)d_in[15];
  const float* Wscs  = (const float*)d_in[16];
  const float* Wscv  = (const float*)d_in[17];

  const long N = in_sizes[0] / C_;
  const long E = in_sizes[3] / B_;

  float* ws    = (float*)d_ws;
  float* s_pre = ws;
  float* v_pre = s_pre + N * C_;
  float* msg_s = v_pre + N * C_ * 3;
  float* msg_v = msg_s + N * C_;

  const int ntiles  = (int)((N + 15) / 16);
  const int eblocks = (int)((E + 7) / 8);

  node_pre_kernel<<<ntiles, 128, 0, stream>>>(fs, fv, W1s, W1v, s_pre, v_pre,
                                              msg_s, msg_v);
  edge_kernel<<<eblocks, 256, 0, stream>>>(emb, sh, eidx, E, Wm1, Wm2, s_pre,
                                           v_pre, msg_s, msg_v);
  node_post_kernel<<<ntiles, 128, 0, stream>>>(fs, fv, attrs, msg_s, msg_v,
                                               W2s, W2v, Tus, Tuv, W3s, W3v,
                                               Wscs, Wscv, (float*)d_out);
}